// SAGE_OneHot_MLP_hetero_42150809043601
// MI455X (gfx1250) — compile-verified
//
#include <hip/hip_runtime.h>

// MI455X / gfx1250. wave32. WMMA fp32 path: V_WMMA_F32_16X16X4_F32.
//
// Roofline: edge aggregation (~2 GB of gather + float-atomic scatter) bounds the
// kernel at ~80us @ 23.3 TB/s; total GEMM work is only ~8.3 GFLOP, so matmuls
// stay in fp32 on the matrix pipe (matches reference precision).
// This revision: (a) LDS-stages the 16-row A tile once per block so 8 waves
// don't re-read it 8x from global (deg scaling folded in at staging time),
// (b) exact-index scatter so no lanes idle on the dominant atomic phase.

typedef float v2f __attribute__((ext_vector_type(2)));
typedef float v8f __attribute__((ext_vector_type(8)));

#define F_IN 100
#define H    128
#define H1   128
#define H2   64
#define BN_EPS 1e-5f

// LDS tile stride: 132 mod 64 == 4 -> 16 rows hit 16 distinct bank groups;
// even stride keeps 8B alignment so paired loads can be ds_load_b64.
#define ASTRIDE 132
#define AMAXK   128

// -------------------- utility: zero a float region --------------------
__global__ void zero_kernel(float* __restrict__ p, long n) {
  long i = (long)blockIdx.x * blockDim.x + threadIdx.x;
  long stride = (long)gridDim.x * blockDim.x;
  for (; i < n; i += stride) p[i] = 0.0f;
}

// -------------------- in-degree --------------------
__global__ void degree_kernel(const int* __restrict__ dst, long E,
                              float* __restrict__ deg) {
  long i = (long)blockIdx.x * blockDim.x + threadIdx.x;
  long stride = (long)gridDim.x * blockDim.x;
  for (; i < E; i += stride) atomicAdd(&deg[dst[i]], 1.0f);
}

// -------------------- scatter-add: s[dst,:] += x[src,:] --------------------
// Exact (edge, feature) decomposition: every lane does real work (F=100 case).
__global__ void scatter_kernel(const int* __restrict__ src,
                               const int* __restrict__ dst, long E,
                               const float* __restrict__ x,
                               float* __restrict__ s, int Fdim) {
  long total = E * (long)Fdim;
  long i = (long)blockIdx.x * blockDim.x + threadIdx.x;
  long stride = (long)gridDim.x * blockDim.x;
  for (; i < total; i += stride) {
    long e = i / Fdim;
    int f  = (int)(i - e * (long)Fdim);
    long sr = (long)src[e];
    long dr = (long)dst[e];
    atomicAdd(&s[dr * Fdim + f], x[sr * Fdim + f]);
  }
}

// -------------------- WMMA fp32 fragment helpers --------------------
// A (16x4 f32): lane&15 = M row; (lane>>4)*2 = K pair base; v0=K, v1=K+1.
// B (4x16 f32): lane&15 = N col; same K pair mapping.
// C/D (16x16 f32): lane&15 = N; M = (lane>>4)*8 + vgpr_index.
__device__ __forceinline__ v8f wmma_f32_k4(v2f a, v2f b, v8f c) {
  return __builtin_amdgcn_wmma_f32_16x16x4_f32(false, a, false, b,
                                               (short)0, c, false, false);
}

// -------------------- SAGE layer GEMM --------------------
// Out[i,:] = relu( (S[i,:]/max(deg[i],1)) @ Wl + bl + X[i,:] @ Wr )
// grid.x = N/16 row tiles; blockDim.x = 32 * (ncols/16); wave w owns columns
// [16w, 16w+16). A tiles (mean, x) staged once in LDS, shared by all waves.
__global__ void sage_gemm_kernel(const float* __restrict__ S,
                                 const float* __restrict__ deg,
                                 const float* __restrict__ X,
                                 const float* __restrict__ Wl,
                                 const float* __restrict__ bl,
                                 const float* __restrict__ Wr,
                                 float* __restrict__ Out,
                                 int K, int ncols) {
  __shared__ float smM[16 * ASTRIDE];   // pre-scaled mean tile
  __shared__ float smX[16 * ASTRIDE];   // x tile

  const long m0 = (long)blockIdx.x << 4;

  // cooperative staging: fold 1/deg into the mean tile here
  for (int idx = threadIdx.x; idx < 16 * K; idx += blockDim.x) {
    int rr = idx / K;
    int kk = idx - rr * K;
    long grow = m0 + rr;
    float d = 1.0f / fmaxf(deg[grow], 1.0f);
    smM[rr * ASTRIDE + kk] = S[grow * (long)K + kk] * d;
    smX[rr * ASTRIDE + kk] = X[grow * (long)K + kk];
  }
  __syncthreads();

  const int lane = threadIdx.x & 31;
  const int wv   = threadIdx.x >> 5;
  const int n0   = wv << 4;
  const int r    = lane & 15;          // A row / B col within tile
  const int kh   = (lane >> 4) << 1;   // K half offset (0 or 2)

  v8f acc = {0.f, 0.f, 0.f, 0.f, 0.f, 0.f, 0.f, 0.f};

  const float* mrow = &smM[r * ASTRIDE + kh];
  const float* xrow = &smX[r * ASTRIDE + kh];

  for (int k0 = 0; k0 < K; k0 += 4) {
    v2f a, b;
    a.x = mrow[k0];
    a.y = mrow[k0 + 1];
    b.x = Wl[(long)(k0 + kh)     * ncols + n0 + r];
    b.y = Wl[(long)(k0 + kh + 1) * ncols + n0 + r];
    acc = wmma_f32_k4(a, b, acc);
  }
  for (int k0 = 0; k0 < K; k0 += 4) {
    v2f a, b;
    a.x = xrow[k0];
    a.y = xrow[k0 + 1];
    b.x = Wr[(long)(k0 + kh)     * ncols + n0 + r];
    b.y = Wr[(long)(k0 + kh + 1) * ncols + n0 + r];
    acc = wmma_f32_k4(a, b, acc);
  }

  const float bias  = bl[n0 + r];
  const int   mbase = (lane >> 4) << 3;
  #pragma unroll
  for (int v = 0; v < 8; ++v) {
    float val = fmaxf(acc[v] + bias, 0.0f);
    Out[(m0 + mbase + v) * (long)ncols + n0 + r] = val;
  }
}

// -------------------- plain GEMM + bias (no activation) --------------------
__global__ void mlp_gemm_kernel(const float* __restrict__ X,
                                const float* __restrict__ Wm,
                                const float* __restrict__ bias,
                                float* __restrict__ Out,
                                int K, int ncols) {
  __shared__ float smA[16 * ASTRIDE];

  const long m0 = (long)blockIdx.x << 4;

  for (int idx = threadIdx.x; idx < 16 * K; idx += blockDim.x) {
    int rr = idx / K;
    int kk = idx - rr * K;
    smA[rr * ASTRIDE + kk] = X[(m0 + rr) * (long)K + kk];
  }
  __syncthreads();

  const int lane = threadIdx.x & 31;
  const int wv   = threadIdx.x >> 5;
  const int n0   = wv << 4;
  const int r    = lane & 15;
  const int kh   = (lane >> 4) << 1;

  v8f acc = {0.f, 0.f, 0.f, 0.f, 0.f, 0.f, 0.f, 0.f};
  const float* arow = &smA[r * ASTRIDE + kh];

  for (int k0 = 0; k0 < K; k0 += 4) {
    v2f a, b;
    a.x = arow[k0];
    a.y = arow[k0 + 1];
    b.x = Wm[(long)(k0 + kh)     * ncols + n0 + r];
    b.y = Wm[(long)(k0 + kh + 1) * ncols + n0 + r];
    acc = wmma_f32_k4(a, b, acc);
  }

  const float bv    = bias[n0 + r];
  const int   mbase = (lane >> 4) << 3;
  #pragma unroll
  for (int v = 0; v < 8; ++v) {
    Out[(m0 + mbase + v) * (long)ncols + n0 + r] = acc[v] + bv;
  }
}

// -------------------- BatchNorm: column sum / sumsq --------------------
__global__ void bn_stats_kernel(const float* __restrict__ Y, long Nrows, int C,
                                float* __restrict__ sum,
                                float* __restrict__ sumsq) {
  const int c = threadIdx.x;
  float s = 0.0f, q = 0.0f;
  for (long rrow = blockIdx.x; rrow < Nrows; rrow += gridDim.x) {
    float v = Y[rrow * C + c];
    s += v;
    q += v * v;
  }
  atomicAdd(&sum[c], s);
  atomicAdd(&sumsq[c], q);
}

__global__ void bn_prep_kernel(const float* __restrict__ sum,
                               const float* __restrict__ sumsq,
                               const float* __restrict__ g,
                               const float* __restrict__ be,
                               float ninv,
                               float* __restrict__ scale,
                               float* __restrict__ shift, int C) {
  int c = threadIdx.x;
  if (c < C) {
    float m   = sum[c] * ninv;
    float var = sumsq[c] * ninv - m * m;
    float sc  = g[c] * rsqrtf(var + BN_EPS);
    scale[c] = sc;
    shift[c] = be[c] - m * sc;
  }
}

// z = relu(y*scale + shift), in place. C is a power of two.
__global__ void bn_apply_relu_kernel(float* __restrict__ Y, long n, int Cmask,
                                     const float* __restrict__ scale,
                                     const float* __restrict__ shift) {
  long i = (long)blockIdx.x * blockDim.x + threadIdx.x;
  long stride = (long)gridDim.x * blockDim.x;
  for (; i < n; i += stride) {
    int c = (int)(i & Cmask);
    Y[i] = fmaxf(Y[i] * scale[c] + shift[c], 0.0f);
  }
}

// -------------------- final head: out[i] = z2[i,:] . W3 + b3 --------------------
__global__ void final_dot_kernel(const float* __restrict__ Z,
                                 const float* __restrict__ W3,
                                 const float* __restrict__ b3,
                                 float* __restrict__ out, long Nrows) {
  const int lane = threadIdx.x & 31;
  const long node = (long)blockIdx.x * (blockDim.x >> 5) + (threadIdx.x >> 5);
  if (node >= Nrows) return;
  const float* zr = Z + node * H2;
  float v = zr[lane] * W3[lane] + zr[lane + 32] * W3[lane + 32];
  #pragma unroll
  for (int off = 16; off > 0; off >>= 1) v += __shfl_xor(v, off, 32);
  if (lane == 0) out[node] = v + b3[0];
}

// -------------------- launch --------------------
extern "C" void kernel_launch(void* const* d_in, const int* in_sizes, int n_in,
                              void* d_out, int out_size, void* d_ws, size_t ws_size,
                              hipStream_t stream) {
  const float* x    = (const float*)d_in[0];
  const int*   eidx = (const int*)d_in[1];
  const float* Wl1  = (const float*)d_in[2];
  const float* bl1  = (const float*)d_in[3];
  const float* Wr1  = (const float*)d_in[4];
  const float* Wl2  = (const float*)d_in[5];
  const float* bl2  = (const float*)d_in[6];
  const float* Wr2  = (const float*)d_in[7];
  const float* W1   = (const float*)d_in[8];
  const float* b1   = (const float*)d_in[9];
  const float* g1   = (const float*)d_in[10];
  const float* be1  = (const float*)d_in[11];
  const float* W2   = (const float*)d_in[12];
  const float* b2   = (const float*)d_in[13];
  const float* g2   = (const float*)d_in[14];
  const float* be2  = (const float*)d_in[15];
  const float* W3   = (const float*)d_in[16];
  const float* b3   = (const float*)d_in[17];

  const long N = (long)in_sizes[0] / F_IN;   // 50000
  const long E = (long)in_sizes[1] / 2;      // 800000
  const int* src = eidx;
  const int* dst = eidx + E;

  float* out = (float*)d_out;      // [N]
  float* h1  = out + N;            // [N,H]
  float* h2  = h1 + N * H;         // [N,H]

  // Workspace layout (floats):
  //   deg[N] | A[N*128] (s1, then y1/z1 in place) | B[N*128] (s2, then y2/z2)
  //   | stats: sum1[128] sq1[128] sum2[64] sq2[64] sc1[128] sh1[128] sc2[64] sh2[64]
  float* ws   = (float*)d_ws;
  float* deg  = ws;
  float* A    = deg + N;
  float* B    = A + N * 128;
  float* st   = B + N * 128;
  float* sum1 = st;        float* sq1 = st + 128;
  float* sum2 = st + 256;  float* sq2 = st + 320;
  float* sc1  = st + 384;  float* sh1 = st + 512;
  float* sc2  = st + 640;  float* sh2 = st + 704;
  const long ws_floats = N + 2 * N * 128 + 768;

  const int rowTiles = (int)(N >> 4);   // 3125

  // 0) zero accumulation buffers + stats (every call: graph-replay safe)
  zero_kernel<<<4096, 256, 0, stream>>>(ws, ws_floats);

  // 1) degrees
  degree_kernel<<<2048, 256, 0, stream>>>(dst, E, deg);

  // 2) layer-1 aggregation + SAGE GEMM -> h1
  scatter_kernel<<<16384, 256, 0, stream>>>(src, dst, E, x, A, F_IN);
  sage_gemm_kernel<<<rowTiles, 32 * (H / 16), 0, stream>>>(
      A, deg, x, Wl1, bl1, Wr1, h1, F_IN, H);

  // 3) layer-2 aggregation + SAGE GEMM -> h2
  scatter_kernel<<<16384, 256, 0, stream>>>(src, dst, E, h1, B, H);
  sage_gemm_kernel<<<rowTiles, 32 * (H / 16), 0, stream>>>(
      B, deg, h1, Wl2, bl2, Wr2, h2, H, H);

  // 4) MLP layer 1: y1 = h2 @ W1 + b1 ; BN ; ReLU   (in buffer A)
  mlp_gemm_kernel<<<rowTiles, 32 * (H1 / 16), 0, stream>>>(h2, W1, b1, A, H, H1);
  bn_stats_kernel<<<512, H1, 0, stream>>>(A, N, H1, sum1, sq1);
  bn_prep_kernel<<<1, H1, 0, stream>>>(sum1, sq1, g1, be1, 1.0f / (float)N,
                                       sc1, sh1, H1);
  bn_apply_relu_kernel<<<4096, 256, 0, stream>>>(A, N * H1, H1 - 1, sc1, sh1);

  // 5) MLP layer 2: y2 = z1 @ W2 + b2 ; BN ; ReLU   (in buffer B)
  mlp_gemm_kernel<<<rowTiles, 32 * (H2 / 16), 0, stream>>>(A, W2, b2, B, H1, H2);
  bn_stats_kernel<<<512, H2, 0, stream>>>(B, N, H2, sum2, sq2);
  bn_prep_kernel<<<1, H2, 0, stream>>>(sum2, sq2, g2, be2, 1.0f / (float)N,
                                       sc2, sh2, H2);
  bn_apply_relu_kernel<<<4096, 256, 0, stream>>>(B, N * H2, H2 - 1, sc2, sh2);

  // 6) head: out = z2 @ W3 + b3
  final_dot_kernel<<<(int)((N + 7) / 8), 256, 0, stream>>>(B, W3, b3, out, N);
}